// AttentionGate_63041529971396
// MI455X (gfx1250) — compile-verified
//
#include <hip/hip_runtime.h>
#include <hip/hip_bf16.h>

typedef __attribute__((ext_vector_type(2))) float v2f;
typedef __attribute__((ext_vector_type(8))) float v8f;

#define BB 8
#define SS 4096
#define KSEL 1024

// ---------------- Kernel 0: lens[b] = S - sum(pad[b,:]) ----------------
__global__ void lens_kernel(const float* __restrict__ pad, int* __restrict__ lens) {
    __shared__ float red[256];
    const int b = blockIdx.x;
    const int tid = threadIdx.x;
    float s = 0.f;
    for (int i = tid; i < SS; i += 256) s += pad[(size_t)b * SS + i];
    red[tid] = s;
    __syncthreads();
    for (int off = 128; off > 0; off >>= 1) {
        if (tid < off) red[tid] += red[tid + off];
        __syncthreads();
    }
    if (tid == 0) lens[b] = SS - (int)(red[0] + 0.5f);
}

// ---------- Kernel 1: scores[b,c] = sum over valid rows r of attn[b,r,c] ----------
// WMMA f32 16x16x4 with A = ones: D[m,n] = sum_k B[k,n] (+C). One wave per 16-col tile.
// Main loop is unmasked (wave-uniform scalar trip count); masked tail handles len % 16.
__global__ void colsum_wmma_kernel(const float* __restrict__ attn,
                                   const int* __restrict__ lens,
                                   float* __restrict__ scores) {
    const int wave = threadIdx.x >> 5;
    const int lane = threadIdx.x & 31;
    const int tile = blockIdx.x * (blockDim.x >> 5) + wave;   // 0 .. B*(S/16)-1
    const int b = tile >> 8;                                  // 256 tiles per batch
    const int tile_c = (tile & 255) << 4;
    const int len = __builtin_amdgcn_readfirstlane(lens[b]);  // wave-uniform -> SGPR
    const int full = len & ~15;                               // rows handled unmasked

    v2f a; a.x = 1.0f; a.y = 1.0f;                            // ones A-matrix (16x4)
    v8f acc0 = {}; v8f acc1 = {}; v8f acc2 = {}; v8f acc3 = {};

    const int rowoff = (lane < 16) ? 0 : 2;                   // B layout: K0/K1 lanes 0-15, K2/K3 lanes 16-31
    const float* base = attn + (size_t)b * SS * SS + (size_t)(tile_c + (lane & 15))
                             + (size_t)rowoff * SS;

    int r0 = 0;
    // ---- hot loop: 8 clause loads + 4 WMMAs, no masking, scalar branch ----
    for (; r0 < full; r0 += 16) {
        const size_t ro = (size_t)r0 * SS;
        v2f b0, b1, b2, b3;
        b0.x = base[ro];                    b0.y = base[ro + SS];
        b1.x = base[ro + 4  * (size_t)SS];  b1.y = base[ro + 5  * (size_t)SS];
        b2.x = base[ro + 8  * (size_t)SS];  b2.y = base[ro + 9  * (size_t)SS];
        b3.x = base[ro + 12 * (size_t)SS];  b3.y = base[ro + 13 * (size_t)SS];
        acc0 = __builtin_amdgcn_wmma_f32_16x16x4_f32(false, a, false, b0, (short)0, acc0, false, false);
        acc1 = __builtin_amdgcn_wmma_f32_16x16x4_f32(false, a, false, b1, (short)0, acc1, false, false);
        acc2 = __builtin_amdgcn_wmma_f32_16x16x4_f32(false, a, false, b2, (short)0, acc2, false, false);
        acc3 = __builtin_amdgcn_wmma_f32_16x16x4_f32(false, a, false, b3, (short)0, acc3, false, false);
    }
    // ---- masked tail (runs whenever len < S; never taken when pad == 0) ----
    for (; r0 < SS; r0 += 16) {
        const int r = r0 + rowoff;
        const size_t ro = (size_t)r0 * SS;
        v2f b0, b1, b2, b3;
        b0.x = (r + 0  < len) ? base[ro]                   : 0.f;
        b0.y = (r + 1  < len) ? base[ro + SS]              : 0.f;
        b1.x = (r + 4  < len) ? base[ro + 4  * (size_t)SS] : 0.f;
        b1.y = (r + 5  < len) ? base[ro + 5  * (size_t)SS] : 0.f;
        b2.x = (r + 8  < len) ? base[ro + 8  * (size_t)SS] : 0.f;
        b2.y = (r + 9  < len) ? base[ro + 9  * (size_t)SS] : 0.f;
        b3.x = (r + 12 < len) ? base[ro + 12 * (size_t)SS] : 0.f;
        b3.y = (r + 13 < len) ? base[ro + 13 * (size_t)SS] : 0.f;
        acc0 = __builtin_amdgcn_wmma_f32_16x16x4_f32(false, a, false, b0, (short)0, acc0, false, false);
        acc1 = __builtin_amdgcn_wmma_f32_16x16x4_f32(false, a, false, b1, (short)0, acc1, false, false);
        acc2 = __builtin_amdgcn_wmma_f32_16x16x4_f32(false, a, false, b2, (short)0, acc2, false, false);
        acc3 = __builtin_amdgcn_wmma_f32_16x16x4_f32(false, a, false, b3, (short)0, acc3, false, false);
    }

    const float sum = acc0[0] + acc1[0] + acc2[0] + acc3[0];  // (M=0, N=lane) for lanes 0-15
    if (lane < 16) scores[(size_t)b * SS + tile_c + lane] = sum;
}

// ---------------- Kernel 2: per-batch top-k + ascending-index compaction ----------------
__global__ void __launch_bounds__(1024) topk_kernel(const float* __restrict__ scores,
                                                    const int* __restrict__ lens,
                                                    float* __restrict__ out) {
    __shared__ unsigned int keys[SS];
    __shared__ unsigned int red[1024];
    const int b = blockIdx.x;
    const int tid = threadIdx.x;
    const int len = lens[b];

    // order-preserving key transform; invalid columns -> 0 (== -inf)
    for (int j = 0; j < 4; ++j) {
        const int i = tid + j * 1024;
        unsigned int key = 0u;
        if (i < len) {
            const unsigned int u = __float_as_uint(scores[(size_t)b * SS + i]);
            key = (u & 0x80000000u) ? ~u : (u | 0x80000000u);
        }
        keys[i] = key;
    }
    __syncthreads();

    // binary search: largest t with count(key >= t) >= KSEL  (t = k-th largest key)
    unsigned int lo = 0u, hi = 0xFFFFFFFFu;
    for (int it = 0; it < 32; ++it) {
        const unsigned int d = hi - lo;                      // uniform across block
        const unsigned int mid = lo + (d >> 1) + (d & 1u);   // upper midpoint, no overflow
        unsigned int cnt = 0u;
        for (int j = 0; j < 4; ++j) cnt += (keys[tid + j * 1024] >= mid) ? 1u : 0u;
        red[tid] = cnt;
        __syncthreads();
        for (int off = 512; off > 0; off >>= 1) {
            if (tid < off) red[tid] += red[tid + off];
            __syncthreads();
        }
        const unsigned int total = red[0];
        __syncthreads();
        if (total >= KSEL) lo = mid; else hi = mid - 1u;
        if (lo >= hi) break;                                 // uniform
    }
    const unsigned int t = lo;

    // count strictly-greater elements
    unsigned int cg = 0u;
    for (int j = 0; j < 4; ++j) cg += (keys[tid + j * 1024] > t) ? 1u : 0u;
    red[tid] = cg;
    __syncthreads();
    for (int off = 512; off > 0; off >>= 1) {
        if (tid < off) red[tid] += red[tid + off];
        __syncthreads();
    }
    const unsigned int cnt_gt = red[0];
    __syncthreads();
    const unsigned int need = KSEL - cnt_gt;                 // equals taken in index order

    // packed (gt<<16 | eq) exclusive scan over 4096 elements, 4 contiguous per thread
    unsigned int packs[4];
    unsigned int run = 0u;
    for (int j = 0; j < 4; ++j) {
        const unsigned int key = keys[4 * tid + j];
        const unsigned int g = (key > t) ? 1u : 0u;
        const unsigned int e = (key == t) ? 1u : 0u;
        packs[j] = run;
        run += (g << 16) | e;
    }
    red[tid] = run;
    __syncthreads();
    for (int off = 1; off < 1024; off <<= 1) {               // Hillis-Steele inclusive scan
        const unsigned int v = (tid >= (unsigned)off) ? red[tid - off] : 0u;
        __syncthreads();
        red[tid] += v;
        __syncthreads();
    }
    const unsigned int excl = red[tid] - run;                // exclusive prefix of thread totals

    float* const out_batch = out + 1;
    float* const out_x     = out + 1 + BB * KSEL;
    float* const out_y     = out + 1 + 2 * BB * KSEL;

    for (int j = 0; j < 4; ++j) {
        const int idx = 4 * tid + j;
        const unsigned int key = keys[idx];
        const unsigned int pre = excl + packs[j];
        const unsigned int gt_before = pre >> 16;
        const unsigned int eq_before = pre & 0xFFFFu;
        const bool g = key > t;
        const bool e = (key == t) && (eq_before < need);
        if (g || e) {
            const unsigned int rank = gt_before + (eq_before < need ? eq_before : need);
            out_x[(size_t)b * KSEL + rank] = (float)idx;
        }
    }
    out_batch[(size_t)b * KSEL + tid] = (float)b;
    out_y[(size_t)b * KSEL + tid] = (float)tid;
    if (b == 0 && tid == 0) out[0] = (float)KSEL;
}

extern "C" void kernel_launch(void* const* d_in, const int* in_sizes, int n_in,
                              void* d_out, int out_size, void* d_ws, size_t ws_size,
                              hipStream_t stream) {
    const float* pad  = (const float*)d_in[0];   // [B,S] f32
    const float* attn = (const float*)d_in[1];   // [B,S,S] f32
    float* out = (float*)d_out;                  // 1 + 3*B*K floats

    float* scores = (float*)d_ws;                                          // B*S floats = 128 KB
    int*   lens   = (int*)((char*)d_ws + (size_t)BB * SS * sizeof(float)); // 8 ints

    lens_kernel<<<BB, 256, 0, stream>>>(pad, lens);
    // B*(S/16) = 2048 column tiles, 8 waves per 256-thread block -> 256 blocks
    colsum_wmma_kernel<<<(BB * (SS / 16)) / 8, 256, 0, stream>>>(attn, lens, scores);
    topk_kernel<<<BB, 1024, 0, stream>>>(scores, lens, out);
}